// SupervisedBCEWithGraphConsistency_72043781423275
// MI455X (gfx1250) — compile-verified
//
#include <hip/hip_runtime.h>
#include <math.h>

typedef __attribute__((ext_vector_type(2))) float v2f;
typedef __attribute__((ext_vector_type(8))) float v8f;

#define GRAPH_WEIGHT 10.0f
#define EPS_W 1e-8f
#define KN2_INVALID 3.0e38f   // forces dist huge -> exp(-dist) == 0 for invalid neighbors

__device__ __forceinline__ float dsigmoid(float x) { return 1.0f / (1.0f + __expf(-x)); }

// ws layout (floats): [0] = bce_sum, [1 .. B] = loss_b, [1+B .. 2B] = num_q
__global__ void zero_ws_kernel(float* ws, int n) {
    int i = blockIdx.x * blockDim.x + threadIdx.x;
    if (i < n) ws[i] = 0.f;
}

__global__ void bce_kernel(const float* __restrict__ logits,
                           const float* __restrict__ tgt,
                           int N, int Nsup, int total, float* __restrict__ ws) {
    int i = blockIdx.x * blockDim.x + threadIdx.x;
    float v = 0.f;
    if (i < total) {
        int b = i / Nsup, n = i - b * Nsup;
        float ls = logits[b * N + n];
        float t  = tgt[i];
        v = fmaxf(ls, 0.f) - ls * t + log1pf(__expf(-fabsf(ls)));
    }
    __shared__ float red[256];
    red[threadIdx.x] = v;
    __syncthreads();
    for (int s = blockDim.x >> 1; s > 0; s >>= 1) {
        if ((int)threadIdx.x < s) red[threadIdx.x] += red[threadIdx.x + s];
        __syncthreads();
    }
    if (threadIdx.x == 0) atomicAdd(&ws[0], red[0]);
}

// One workgroup per (b, qb). 256 threads = 8 waves; wave w owns query rows
// [w*16, w*16+16). 64 neighbor tiles of 16; each tile's 16x16 q·k cross terms
// come from one v_wmma_f32_16x16x4_f32 (K=2 padded to 4 with zeros).
__global__ void graph_kernel(const float* __restrict__ logits,
                             const float* __restrict__ pos,
                             const unsigned char* __restrict__ sup_mask,
                             const unsigned char* __restrict__ ign_mask,
                             const int* __restrict__ kv_idx,
                             const int* __restrict__ kv_num,
                             int B, int N, int NB, int BS, int K,
                             float* __restrict__ ws) {
    const int b   = blockIdx.x / NB;
    const int qb  = blockIdx.x - b * NB;
    const int KN  = K * BS;
    const int tid = threadIdx.x;
    const int lane = tid & 31;
    const int wave = tid >> 5;
    const int lc   = lane & 15;      // column within 16-lane half
    const int half = lane >> 4;      // 0: rows r, 1: rows r+8

    extern __shared__ char smem_raw[];
    float4* s_k4   = (float4*)smem_raw;            // {kx, ky, kn2', kp} per neighbor
    int*    s_knode = (int*)(s_k4 + KN);
    float*  s_qx   = (float*)(s_knode + KN);
    float*  s_qy   = s_qx + BS;
    float*  s_qp   = s_qy + BS;
    float*  s_quv  = s_qp + BS;
    int*    s_hask = (int*)(s_quv + BS);

    if (tid == 0) *s_hask = 0;
    __syncthreads();

    // ---- stage neighbor block (K*BS nodes) into LDS ----
    const int nblk = kv_num[b * NB + qb];
    int any_valid = 0;
    for (int n = tid; n < KN; n += blockDim.x) {
        int k  = n / BS;
        int j  = n - k * BS;
        int kb = kv_idx[(b * NB + qb) * K + k];
        int node = kb * BS + j;
        float kx = pos[(b * N + node) * 2 + 0];
        float ky = pos[(b * N + node) * 2 + 1];
        int kval = (k < nblk) && (ign_mask[b * N + node] == 0);
        float4 kd;
        kd.x = kx;
        kd.y = ky;
        kd.z = kval ? (kx * kx + ky * ky) : KN2_INVALID;
        kd.w = dsigmoid(logits[b * N + node]);
        s_k4[n] = kd;
        s_knode[n] = node;
        any_valid |= kval;
    }
    // ---- stage query block (BS nodes) ----
    for (int n = tid; n < BS; n += blockDim.x) {
        int node = qb * BS + n;
        s_qx[n] = pos[(b * N + node) * 2 + 0];
        s_qy[n] = pos[(b * N + node) * 2 + 1];
        s_qp[n] = dsigmoid(logits[b * N + node]);
        s_quv[n] = (ign_mask[b * N + node] == 0 && sup_mask[b * N + node] == 0) ? 1.f : 0.f;
    }
    if (any_valid) atomicOr(s_hask, 1);
    __syncthreads();

    // ---- A matrix: 16x4 fp32 (K=0,1 real, K=2,3 zero) ----
    // ISA layout: VGPR0 = K0 (lanes 0-15) / K2 (lanes 16-31); VGPR1 = K1 / K3.
    const int m0 = wave * 16;
    const int qi = m0 + lc;
    v2f avec;
    avec.x = (half == 0) ? s_qx[qi] : 0.f;
    avec.y = (half == 0) ? s_qy[qi] : 0.f;

    float qn2r[8];
    int   qnoder[8];
#pragma unroll
    for (int r = 0; r < 8; r++) {
        int m = m0 + r + half * 8;
        float qx = s_qx[m], qy = s_qy[m];
        qn2r[r]   = qx * qx + qy * qy;
        qnoder[r] = qb * BS + m;
    }

    float wsum[8], wk[8];
#pragma unroll
    for (int r = 0; r < 8; r++) { wsum[r] = 0.f; wk[r] = 0.f; }

    const int ntiles = KN >> 4;
#pragma unroll 2
    for (int t = 0; t < ntiles; t++) {
        int col = t * 16 + lc;
        float4 kd = s_k4[col];        // one ds_load_b128
        int    knd = s_knode[col];    // one ds_load_b32

        // B matrix 4x16: VGPR0 = K0 row (lanes 0-15) / K2 zeros; VGPR1 = K1 / K3.
        v2f bvec;
        bvec.x = (half == 0) ? kd.x : 0.f;
        bvec.y = (half == 0) ? kd.y : 0.f;

        v8f c = {};
        c = __builtin_amdgcn_wmma_f32_16x16x4_f32(
                false, avec, false, bvec, (short)0, c, false, false);
        // c[r]: row = m0 + r + half*8, col = t*16 + lc

#pragma unroll
        for (int r = 0; r < 8; r++) {
            float d2   = qn2r[r] + kd.z - 2.f * c[r];
            float dist = __builtin_amdgcn_sqrtf(fmaxf(d2, 0.f)); // raw v_sqrt_f32
            float w    = __expf(-dist);                          // SIGMA == 1
            w = (knd == qnoder[r]) ? 0.f : w;                    // self mask
            wsum[r] += w;
            wk[r]   += w * kd.w;
        }
    }

    // ---- reduce the 16 columns of each half-wave (xor 1,2,4,8 stays in-half) ----
#pragma unroll
    for (int r = 0; r < 8; r++) {
#pragma unroll
        for (int off = 1; off < 16; off <<= 1) {
            wsum[r] += __shfl_xor(wsum[r], off, 32);
            wk[r]   += __shfl_xor(wk[r],   off, 32);
        }
    }

    const int hask = *s_hask;
    if (lc == 0) {   // lane 0 finalizes rows m0+0..7, lane 16 rows m0+8..15
        float loss = 0.f, cnt = 0.f;
#pragma unroll
        for (int r = 0; r < 8; r++) {
            int m = m0 + r + half * 8;
            float kmean = wk[r] / (wsum[r] + EPS_W);
            float diff  = s_qp[m] - kmean;
            float use   = (hask && s_quv[m] > 0.5f) ? 1.f : 0.f;
            loss += use * diff * diff;
            cnt  += use;
        }
        atomicAdd(&ws[1 + b], loss);
        atomicAdd(&ws[1 + B + b], cnt);
    }
}

__global__ void finalize_kernel(const float* __restrict__ ws, float* __restrict__ out,
                                int B, int total_sup) {
    float loss_sup = ws[0] / (float)total_sup;
    float s = 0.f;
    int nv = 0;
    for (int b = 0; b < B; b++) {
        float nq = ws[1 + B + b];
        if (nq > 0.5f) { s += ws[1 + b] / fmaxf(nq, 1.f); nv++; }
    }
    if (nv < 1) nv = 1;
    out[0] = loss_sup + GRAPH_WEIGHT * (s / (float)nv);
}

extern "C" void kernel_launch(void* const* d_in, const int* in_sizes, int n_in,
                              void* d_out, int out_size, void* d_ws, size_t ws_size,
                              hipStream_t stream) {
    const float* logits = (const float*)d_in[0];
    const float* tgt    = (const float*)d_in[1];
    const float* pos    = (const float*)d_in[2];
    const unsigned char* supm = (const unsigned char*)d_in[3];
    const unsigned char* ignm = (const unsigned char*)d_in[4];
    const int* kv_idx = (const int*)d_in[5];
    const int* kv_num = (const int*)d_in[6];

    const int B    = 2;
    const int N    = in_sizes[0] / B;
    const int Nsup = in_sizes[1] / B;
    const int NB   = in_sizes[6] / B;
    const int K    = in_sizes[5] / in_sizes[6];
    const int BS   = N / NB;
    const int KN   = K * BS;

    float* ws = (float*)d_ws;
    const int nws = 1 + 2 * B;
    zero_ws_kernel<<<1, 32, 0, stream>>>(ws, nws);

    const int total_sup = B * Nsup;
    bce_kernel<<<(total_sup + 255) / 256, 256, 0, stream>>>(logits, tgt, N, Nsup, total_sup, ws);

    // shared: float4[KN] + int[KN] + 4*float[BS] + int
    size_t shmem = (size_t)KN * 16 + (size_t)KN * 4 + (size_t)BS * 4 * 4 + 8;
    graph_kernel<<<B * NB, 256, shmem, stream>>>(logits, pos, supm, ignm,
                                                 kv_idx, kv_num,
                                                 B, N, NB, BS, K, ws);

    finalize_kernel<<<1, 1, 0, stream>>>(ws, (float*)d_out, B, total_sup);
}